// GraphAttentionLayer_5480378270303
// MI455X (gfx1250) — compile-verified
//
#include <hip/hip_runtime.h>
#include <cstddef>
#include <cstdint>

typedef __attribute__((ext_vector_type(2))) float v2f;
typedef __attribute__((ext_vector_type(8))) float v8f;

#define ALPHA    0.2f
#define NEG_BIG  -9000000000000000.0f

constexpr int Bn  = 32;
constexpr int Nn  = 1024;
constexpr int Fo  = 64;
constexpr int TJ  = 128;   // j-tile width for fused softmax+GEMM
constexpr int WV  = 2;     // waves per block in kernel 2
constexpr int PAD = 4;     // LDS row pad; matches TDM pad_amount = 4 DWORDs

#if __has_builtin(__builtin_amdgcn_tensor_load_to_lds)
#define HAVE_TDM 1
typedef unsigned int u32x4 __attribute__((ext_vector_type(4)));
typedef int          i32x4 __attribute__((ext_vector_type(4)));
typedef int          i32x8 __attribute__((ext_vector_type(8)));
#else
#define HAVE_TDM 0
#endif

// ---------------------------------------------------------------------------
// Kernel 1: h = x @ W ; f = h @ a1 ; g = h @ a2     (tiny: 0.27 GFLOP)
// block = 256 threads = 4 rows x 64 lanes(=Fout)
// ---------------------------------------------------------------------------
__global__ __launch_bounds__(256) void gat_hfg(
    const float* __restrict__ x, const float* __restrict__ W,
    const float* __restrict__ a, float* __restrict__ h,
    float* __restrict__ f, float* __restrict__ g)
{
  __shared__ float sW[Fo][Fo];
  __shared__ float sA[2 * Fo];
  __shared__ float sX[4][Fo];
  __shared__ float sPart[8][2];

  const int tid = threadIdx.x;
  const int rl  = tid >> 6;       // row within block (0..3)
  const int o   = tid & 63;       // output feature
  const size_t row = (size_t)blockIdx.x * 4 + rl;   // global row in [0, B*N)

  for (int t = tid; t < Fo * Fo; t += 256) sW[t >> 6][t & 63] = W[t];
  if (tid < 2 * Fo) sA[tid] = a[tid];
  sX[rl][o] = x[row * Fo + o];
  __syncthreads();

  float acc = 0.f;
  #pragma unroll
  for (int k = 0; k < Fo; ++k) acc = fmaf(sX[rl][k], sW[k][o], acc);
  h[row * Fo + o] = acc;

  // deterministic reduction for f,g: shuffle tree within wave32, combine 2 waves
  float pf = acc * sA[o];
  float pg = acc * sA[Fo + o];
  #pragma unroll
  for (int off = 16; off > 0; off >>= 1) {
    pf += __shfl_down(pf, off);
    pg += __shfl_down(pg, off);
  }
  if ((tid & 31) == 0) { sPart[tid >> 5][0] = pf; sPart[tid >> 5][1] = pg; }
  __syncthreads();
  if (o == 0) {
    f[row] = sPart[2 * rl][0] + sPart[2 * rl + 1][0];
    g[row] = sPart[2 * rl][1] + sPart[2 * rl + 1][1];
  }
}

// ---------------------------------------------------------------------------
// Kernel 2: fused masked-softmax + (att @ h) + ELU, flash-attention style.
// block = 64 threads = 2 waves; each wave owns a 16-row output tile (16x64).
// h tile staged to LDS via the Tensor Data Mover (TDM pad feature reproduces
// the [TJ][Fo+4] padded layout). Accumulator = 4x 16x16 f32 WMMA C tiles.
// ---------------------------------------------------------------------------
__global__ __launch_bounds__(WV * 32) void gat_attn(
    const float* __restrict__ h, const float* __restrict__ f,
    const float* __restrict__ g, const int* __restrict__ adj,
    float* __restrict__ out)
{
  __shared__ __align__(16) float sH[TJ][Fo + PAD];   // staged h chunk (shared by both waves)
  __shared__ __align__(16) float sGt[TJ];            // staged g chunk
  __shared__ __align__(16) float sP[WV][16][TJ];     // per-wave e -> p buffer
  __shared__ float sScale[WV][16];
  __shared__ float sLfin[WV][16];

  const int b    = blockIdx.y;
  const int tile = blockIdx.x;                 // 32 tiles of 32 rows
  const int tid  = threadIdx.x;
  const int wv   = tid >> 5;
  const int lane = tid & 31;
  const int i0   = tile * (WV * 16) + wv * 16; // wave's first output row

  // phase-A mapping: 2 lanes per row, 64 cols each
  const int r   = lane >> 1;
  const int hh  = lane & 1;
  // WMMA lane mapping
  const int m16 = lane & 15;
  const int hi  = lane >> 4;                   // 0: K/M low half, 1: high half
  const int bk  = hi ? 2 : 0;                  // A/B VGPR0 holds K = bk

  const float fr = f[b * Nn + i0 + r];
  const int* adjRow = adj + ((size_t)b * Nn + (i0 + r)) * Nn;
  const float* hB = h + (size_t)b * Nn * Fo;

  v8f acc0 = {}, acc1 = {}, acc2 = {}, acc3 = {};
  float m_run = NEG_BIG;
  float l_run = 0.f;

  for (int j0 = 0; j0 < Nn; j0 += TJ) {
    __syncthreads();  // previous tile's WMMA LDS reads done before restaging

#if HAVE_TDM
    // ---- stage h[j0:j0+TJ, 0:64] into sH with one TDM 2D tile load (wave 0)
    if (wv == 0) {
      const uint64_t ga  = (uint64_t)(uintptr_t)(hB + (size_t)j0 * Fo);
      const uint32_t lds = (uint32_t)(uintptr_t)&sH[0][0];  // low 32b = LDS offset
      u32x4 g0;
      g0[0] = 1u;                                   // count=1 (valid user D#)
      g0[1] = lds;                                  // lds_addr
      g0[2] = (uint32_t)ga;                         // global_addr[31:0]
      g0[3] = (uint32_t)((ga >> 32) & 0x1FFFFFFu)   // global_addr[56:32]
              | 0x80000000u;                        // type=2 ("image")
      i32x8 g1;
      g1[0] = (int)((2u << 16)      // data_size = 4 bytes
              | (1u << 20)          // pad_enable
              | (5u << 22)          // pad_interval: 2<<5 = 64 DWORDs per row
              | (3u << 25));        // pad_amount: 4 DWORDs (= PAD)
      g1[1] = (int)((uint32_t)Fo << 16);    // tensor_dim0 = 64 (bits 63:48)
      g1[2] = (int)((uint32_t)Nn << 16);    // dim0 hi=0 | tensor_dim1 = 1024
      g1[3] = (int)((uint32_t)Fo << 16);    // dim1 hi=0 | tile_dim0 = 64
      g1[4] = (int)TJ;                      // tile_dim1 = 128, tile_dim2 = 0
      g1[5] = (int)Fo;                      // tensor_dim0_stride = 64
      g1[6] = 0;                            // stride hi / dim1_stride lo
      g1[7] = 0;
      const i32x4 gz4 = {0, 0, 0, 0};       // groups 2/3 unused (2D tensor)
      const i32x8 gz8 = {0, 0, 0, 0, 0, 0, 0, 0};
      __builtin_amdgcn_tensor_load_to_lds(g0, g1, gz4, gz4, gz8, 0);
      __builtin_amdgcn_s_wait_tensorcnt(0);
    }
#else
    for (int t = tid; t < TJ * Fo / 4; t += WV * 32) {
      float4 v = ((const float4*)(hB + (size_t)j0 * Fo))[t];
      int rr = t / (Fo / 4);
      int cc = (t % (Fo / 4)) * 4;
      sH[rr][cc + 0] = v.x; sH[rr][cc + 1] = v.y;
      sH[rr][cc + 2] = v.z; sH[rr][cc + 3] = v.w;
    }
#endif
    for (int t = tid; t < TJ; t += WV * 32) sGt[t] = g[b * Nn + j0 + t];
    __syncthreads();

    if (j0 + TJ < Nn) __builtin_prefetch(adjRow + j0 + TJ, 0, 0);

    // ---- phase A: masked leaky-relu e into sP; per-row running max
    const int cb = hh * (TJ / 2);
    float mloc = NEG_BIG;
    for (int jj = 0; jj < TJ / 2; jj += 4) {
      int4   ad = *(const int4*)(adjRow + j0 + cb + jj);
      float4 gv = *(const float4*)(&sGt[cb + jj]);
      float e0 = fr + gv.x; e0 = (e0 > 0.f) ? e0 : ALPHA * e0; e0 = (ad.x > 0) ? e0 : NEG_BIG;
      float e1 = fr + gv.y; e1 = (e1 > 0.f) ? e1 : ALPHA * e1; e1 = (ad.y > 0) ? e1 : NEG_BIG;
      float e2 = fr + gv.z; e2 = (e2 > 0.f) ? e2 : ALPHA * e2; e2 = (ad.z > 0) ? e2 : NEG_BIG;
      float e3 = fr + gv.w; e3 = (e3 > 0.f) ? e3 : ALPHA * e3; e3 = (ad.w > 0) ? e3 : NEG_BIG;
      *(float4*)(&sP[wv][r][cb + jj]) = make_float4(e0, e1, e2, e3);
      mloc = fmaxf(mloc, fmaxf(fmaxf(e0, e1), fmaxf(e2, e3)));
    }
    mloc = fmaxf(mloc, __shfl_xor(mloc, 1));     // pair-combine (2 lanes per row)
    const float m_new = fmaxf(m_run, mloc);
    const float scale = __expf(m_run - m_new);   // online-softmax rescale factor
    m_run = m_new;

    // ---- phase B: p = exp(e - m_new), row sum (same-lane LDS, no barrier needed)
    float sloc = 0.f;
    for (int jj = 0; jj < TJ / 2; jj += 4) {
      float4 ev = *(const float4*)(&sP[wv][r][cb + jj]);
      float p0 = __expf(ev.x - m_new);
      float p1 = __expf(ev.y - m_new);
      float p2 = __expf(ev.z - m_new);
      float p3 = __expf(ev.w - m_new);
      *(float4*)(&sP[wv][r][cb + jj]) = make_float4(p0, p1, p2, p3);
      sloc += p0 + p1 + p2 + p3;
    }
    sloc += __shfl_xor(sloc, 1);
    l_run = l_run * scale + sloc;
    if (hh == 0) sScale[wv][r] = scale;
    __syncthreads();   // sScale + sP visible cross-lane

    // ---- rescale accumulators by per-row factor
    float sc[8];
    #pragma unroll
    for (int v = 0; v < 8; ++v) sc[v] = sScale[wv][hi * 8 + v];
    #pragma unroll
    for (int v = 0; v < 8; ++v) {
      acc0[v] *= sc[v]; acc1[v] *= sc[v]; acc2[v] *= sc[v]; acc3[v] *= sc[v];
    }

    // ---- WMMA K-loop: acc(16x64) += P(16xTJ) * H(TJx64), K step = 4 (f32)
    for (int jj = 0; jj < TJ; jj += 4) {
      v2f A;
      A[0] = sP[wv][m16][jj + bk];
      A[1] = sP[wv][m16][jj + bk + 1];
      v2f B0, B1, B2, B3;
      B0[0] = sH[jj + bk][ 0 + m16]; B0[1] = sH[jj + bk + 1][ 0 + m16];
      B1[0] = sH[jj + bk][16 + m16]; B1[1] = sH[jj + bk + 1][16 + m16];
      B2[0] = sH[jj + bk][32 + m16]; B2[1] = sH[jj + bk + 1][32 + m16];
      B3[0] = sH[jj + bk][48 + m16]; B3[1] = sH[jj + bk + 1][48 + m16];
      acc0 = __builtin_amdgcn_wmma_f32_16x16x4_f32(false, A, false, B0, (short)0, acc0, false, false);
      acc1 = __builtin_amdgcn_wmma_f32_16x16x4_f32(false, A, false, B1, (short)0, acc1, false, false);
      acc2 = __builtin_amdgcn_wmma_f32_16x16x4_f32(false, A, false, B2, (short)0, acc2, false, false);
      acc3 = __builtin_amdgcn_wmma_f32_16x16x4_f32(false, A, false, B3, (short)0, acc3, false, false);
    }
  }

  // ---- finalize: divide by softmax denominator, apply ELU, store
  if (hh == 0) sLfin[wv][r] = l_run;
  __syncthreads();

  float* outB = out + ((size_t)b * Nn + i0) * Fo;
  #pragma unroll
  for (int v = 0; v < 8; ++v) {
    const float linv = 1.0f / sLfin[wv][hi * 8 + v];
    const int orow = hi * 8 + v;
    float p0 = acc0[v] * linv, p1 = acc1[v] * linv;
    float p2 = acc2[v] * linv, p3 = acc3[v] * linv;
    outB[orow * Fo +  0 + m16] = (p0 > 0.f) ? p0 : __expf(p0) - 1.f;
    outB[orow * Fo + 16 + m16] = (p1 > 0.f) ? p1 : __expf(p1) - 1.f;
    outB[orow * Fo + 32 + m16] = (p2 > 0.f) ? p2 : __expf(p2) - 1.f;
    outB[orow * Fo + 48 + m16] = (p3 > 0.f) ? p3 : __expf(p3) - 1.f;
  }
}

// ---------------------------------------------------------------------------
extern "C" void kernel_launch(void* const* d_in, const int* in_sizes, int n_in,
                              void* d_out, int out_size, void* d_ws, size_t ws_size,
                              hipStream_t stream) {
  (void)in_sizes; (void)n_in; (void)out_size; (void)ws_size;
  const float* x   = (const float*)d_in[0];   // [32,1024,64] f32
  const int*   adj = (const int*)  d_in[1];   // [32,1024,1024] i32
  const float* W   = (const float*)d_in[2];   // [64,64] f32
  const float* a   = (const float*)d_in[3];   // [128,1] f32
  float* out = (float*)d_out;                 // [32,1024,64] f32

  // workspace carve: h (8 MB), f (128 KB), g (128 KB)
  float* h = (float*)d_ws;
  float* f = h + (size_t)Bn * Nn * Fo;
  float* g = f + (size_t)Bn * Nn;

  gat_hfg<<<dim3(Bn * Nn / 4), 256, 0, stream>>>(x, W, a, h, f, g);
  gat_attn<<<dim3(Nn / (WV * 16), Bn), WV * 32, 0, stream>>>(h, f, g, adj, out);
}